// VolumeExplicit_29257317220866
// MI455X (gfx1250) — compile-verified
//
#include <hip/hip_runtime.h>

// VolumeExplicit trilinear grid-sample (align_corners=False, padding_mode='zeros')
// out[b,p] = 100 * trilinear(volume, x[b,p,:] / EXTENT);  EXTENT=0.5, RES=256.
//
// MI455X reasoning: ~18 MFLOP vs ~72 MB traffic -> pure gather-bound.
// 64 MB volume is L2-resident (192 MB L2); limiter = per-lane VMEM gather
// requests. So: fuse each x-neighbor pair into ONE b64 gather (8 -> 4 requests),
// wave32 + tiny VGPR footprint for deep latency hiding, NT hints on the
// streamed coords/output so L2 capacity stays with the volume.

#define RES   256
#define RESM1 255

struct __attribute__((packed)) f3 { float x, y, z; };

// 2-float vector with alignment relaxed to 4 bytes: x0 may be odd, so the
// pair load is only DWORD-aligned. gfx1250 handles unaligned global loads;
// this lowers to a single global_load_b64.
typedef float f2v   __attribute__((ext_vector_type(2)));
typedef f2v  f2v_u4 __attribute__((aligned(4)));

__global__ __launch_bounds__(256) void VolumeExplicit_29257317220866_kernel(
    const float* __restrict__ xin,      // [B, N, 3] flattened
    const float* __restrict__ vol,      // [256,256,256] flattened (z,y,x)
    float* __restrict__ out,            // [B, N] flattened
    int n)                              // total points
{
    int i = blockIdx.x * blockDim.x + threadIdx.x;
    if (i >= n) return;

    // --- coords: streamed once, non-temporal to protect L2 for the volume
    const f3* __restrict__ cp = (const f3*)(xin) + i;
    float cx = __builtin_nontemporal_load(&cp->x);
    float cy = __builtin_nontemporal_load(&cp->y);
    float cz = __builtin_nontemporal_load(&cp->z);

    // coords = x / EXTENT = 2*x (exact);  ix = ((g+1)*256 - 1)*0.5 = g*128 + 127.5
    float ix = fmaf(2.0f * cx, 128.0f, 127.5f);
    float iy = fmaf(2.0f * cy, 128.0f, 127.5f);
    float iz = fmaf(2.0f * cz, 128.0f, 127.5f);

    float fx = floorf(ix), fy = floorf(iy), fz = floorf(iz);
    float tx = ix - fx,    ty = iy - fy,    tz = iz - fz;

    int x0 = (int)fx, y0 = (int)fy, z0 = (int)fz;
    int x1 = x0 + 1,  y1 = y0 + 1,  z1 = z0 + 1;

    // padding_mode='zeros': fold per-corner validity into the 1-D weights,
    // then gather at clamped indices (weight==0 kills any clamped value).
    float wx0 = (x0 >= 0 && x0 <= RESM1) ? (1.0f - tx) : 0.0f;
    float wx1 = (x1 >= 0 && x1 <= RESM1) ? tx          : 0.0f;
    float wy0 = (y0 >= 0 && y0 <= RESM1) ? (1.0f - ty) : 0.0f;
    float wy1 = (y1 >= 0 && y1 <= RESM1) ? ty          : 0.0f;
    float wz0 = (z0 >= 0 && z0 <= RESM1) ? (1.0f - tz) : 0.0f;
    float wz1 = (z1 >= 0 && z1 <= RESM1) ? tz          : 0.0f;

    int xc0 = min(max(x0, 0), RESM1), xc1 = min(max(x1, 0), RESM1);
    int yc0 = min(max(y0, 0), RESM1), yc1 = min(max(y1, 0), RESM1);
    int zc0 = min(max(z0, 0), RESM1), zc1 = min(max(z1, 0), RESM1);

    // x-pair fusion: one b64 load of {vol[pb], vol[pb+1]} per row.
    // pb = min(xc0, 254) keeps the hi element in-bounds at the x=255 edge.
    // Map (xc0, xc1) picks into adjusted lo/hi weights once per thread:
    //   normal:      xc0->lo, xc1->hi  => (axlo, axhi) = (wx0, wx1)
    //   left pad:    xc0=xc1=0  -> lo  => (wx0+wx1, 0)   [wx0 already 0]
    //   right clamp: xc0=xc1=255-> hi  => (0, wx0+wx1)   [wx1 already 0]
    int  pb = min(xc0, RESM1 - 1);
    bool s0 = (xc0 != pb);              // xc0 selects hi
    bool s1 = (xc1 != pb);              // xc1 selects hi (normal case)
    float axlo = (s0 ? 0.0f : wx0) + (s1 ? 0.0f : wx1);
    float axhi = (s0 ? wx0 : 0.0f) + (s1 ? wx1 : 0.0f);

    // row bases: idx = (z*256 + y)*256 + pb
    int b00 = (((zc0 << 8) + yc0) << 8) + pb;
    int b01 = (((zc0 << 8) + yc1) << 8) + pb;
    int b10 = (((zc1 << 8) + yc0) << 8) + pb;
    int b11 = (((zc1 << 8) + yc1) << 8) + pb;

    // 4 independent b64 gathers (L2-resident volume)
    f2v p00 = *(const f2v_u4*)(vol + b00);
    f2v p01 = *(const f2v_u4*)(vol + b01);
    f2v p10 = *(const f2v_u4*)(vol + b10);
    f2v p11 = *(const f2v_u4*)(vol + b11);

    // blend: x within each pair, then y, then z (all FMA chains)
    float r00 = fmaf(p00.y, axhi, p00.x * axlo);
    float r01 = fmaf(p01.y, axhi, p01.x * axlo);
    float r10 = fmaf(p10.y, axhi, p10.x * axlo);
    float r11 = fmaf(p11.y, axhi, p11.x * axlo);

    float rz0 = fmaf(wy1, r01, wy0 * r00);
    float rz1 = fmaf(wy1, r11, wy0 * r10);
    float r   = fmaf(wz1, rz1, wz0 * rz0);

    __builtin_nontemporal_store(100.0f * r, out + i);
}

extern "C" void kernel_launch(void* const* d_in, const int* in_sizes, int n_in,
                              void* d_out, int out_size, void* d_ws, size_t ws_size,
                              hipStream_t stream) {
    const float* x   = (const float*)d_in[0];   // [8, 65536, 3] f32
    const float* vol = (const float*)d_in[1];   // [256,256,256] f32
    float* out = (float*)d_out;                 // [8, 65536] f32

    int n = out_size;                           // 524288 points
    const int threads = 256;                    // 8 wave32s per block
    int blocks = (n + threads - 1) / threads;
    VolumeExplicit_29257317220866_kernel<<<blocks, threads, 0, stream>>>(x, vol, out, n);
}